// L2SuperKnotDistiller_56126632624492
// MI455X (gfx1250) — compile-verified
//
#include <hip/hip_runtime.h>
#include <math.h>

// Problem constants (fixed by setup_inputs in the reference)
#define BATCH 2
#define NCURVE 32
#define SPTS 96
#define SEGP 96      // 95 real segments + 1 zero pad
#define LATENT 256
#define EPSV 1e-8f
#define FOUR_PI 12.566370614359172f

typedef __attribute__((ext_vector_type(2))) float v2f;
typedef __attribute__((ext_vector_type(4))) float v4f;
typedef __attribute__((ext_vector_type(8))) float v8f;

// Workspace layout (floats):
//  ops: [BATCH*NCURVE][SEGP][32]
//       per segment: [0..7]  A_num = {c.x,c.y,c.z,d.x,d.y,d.z,0,0}
//                    [8..15] A_dst = {-2m.x,-2m.y,-2m.z,|m|^2,1,0,0,0}
//                    [16..23] B_num = {d.x,d.y,d.z,c.x,c.y,c.z,0,0}
//                    [24..31] B_dst = {m.x,m.y,m.z,1,|m|^2,0,0,0}
//  lk:  [BATCH][NCURVE][NCURVE]
#define OPS_PER_CURVE (SEGP * 32)
#define OPS_TOTAL (BATCH * NCURVE * OPS_PER_CURVE)

// ---------------------------------------------------------------------------
// Kernel 1: build GEMM operands from quaternions
// grid = BATCH*NCURVE blocks, SEGP threads
// ---------------------------------------------------------------------------
__global__ void prep_kernel(const float* __restrict__ q, float* __restrict__ ops) {
  const int curve = blockIdx.x;          // b*NCURVE + n
  const int s = threadIdx.x;             // segment index 0..95
  float* o = ops + (size_t)curve * OPS_PER_CURVE + (size_t)s * 32;

  if (s >= SPTS - 1) {
    // zero pad: contributes num=0 and denom=eps -> 0 contribution
    #pragma unroll
    for (int k = 0; k < 8; ++k) ((v4f*)o)[k] = (v4f){0.f, 0.f, 0.f, 0.f};
    return;
  }

  const float* qp = q + ((size_t)curve * SPTS + s) * 4;
  float a0 = qp[0], a1 = qp[1], a2 = qp[2], a3 = qp[3];
  float b0 = qp[4], b1 = qp[5], b2 = qp[6], b3 = qp[7];
  float inva = 1.f / (sqrtf(a0 * a0 + a1 * a1 + a2 * a2 + a3 * a3) + EPSV);
  float invb = 1.f / (sqrtf(b0 * b0 + b1 * b1 + b2 * b2 + b3 * b3) + EPSV);
  // r = normalized xyz (components 1..3)
  float r0x = a1 * inva, r0y = a2 * inva, r0z = a3 * inva;
  float r1x = b1 * invb, r1y = b2 * invb, r1z = b3 * invb;

  float mx = 0.5f * (r0x + r1x), my = 0.5f * (r0y + r1y), mz = 0.5f * (r0z + r1z);
  float dx = r1x - r0x, dy = r1y - r0y, dz = r1z - r0z;
  // c = m x d
  float cx = my * dz - mz * dy;
  float cy = mz * dx - mx * dz;
  float cz = mx * dy - my * dx;
  float mn = mx * mx + my * my + mz * mz;

  ((v4f*)o)[0] = (v4f){cx, cy, cz, dx};                       // A_num[0..3]
  ((v4f*)o)[1] = (v4f){dy, dz, 0.f, 0.f};                     // A_num[4..7]
  ((v4f*)o)[2] = (v4f){-2.f * mx, -2.f * my, -2.f * mz, mn};  // A_dst[0..3]
  ((v4f*)o)[3] = (v4f){1.f, 0.f, 0.f, 0.f};                   // A_dst[4..7]
  ((v4f*)o)[4] = (v4f){dx, dy, dz, cx};                       // B_num[0..3]
  ((v4f*)o)[5] = (v4f){cy, cz, 0.f, 0.f};                     // B_num[4..7]
  ((v4f*)o)[6] = (v4f){mx, my, mz, 1.f};                      // B_dst[0..3]
  ((v4f*)o)[7] = (v4f){mn, 0.f, 0.f, 0.f};                    // B_dst[4..7]
}

// ---------------------------------------------------------------------------
// Kernel 2: per curve-pair Gauss linking sum via f32 WMMA (16x16x4)
// grid = BATCH * 528 blocks (upper triangle incl. diagonal), 256 threads
// ---------------------------------------------------------------------------
__global__ void __launch_bounds__(256)
pair_kernel(const float* __restrict__ ops, float* __restrict__ lk) {
  const int b = blockIdx.x / 528;
  int rem = blockIdx.x % 528;
  int i = 0;
  while (rem >= NCURVE - i) { rem -= NCURVE - i; ++i; }
  const int j = i + rem;

  if (i == j) {                       // diagonal is zeroed by the reference
    if (threadIdx.x == 0) lk[((size_t)b * NCURVE + i) * NCURVE + i] = 0.f;
    return;                           // uniform exit before any barrier
  }

  __shared__ float sA[SEGP * 16];     // A_num8 + A_dst8 of curve i
  __shared__ float sB[SEGP * 16];     // B_num8 + B_dst8 of curve j
  __shared__ float sRed[256];

  const float* gA = ops + (size_t)(b * NCURVE + i) * OPS_PER_CURVE;        // fields 0..15
  const float* gB = ops + (size_t)(b * NCURVE + j) * OPS_PER_CURVE + 16;   // fields 16..31

  // cooperative stage: 384 float4 per array
  for (int t = threadIdx.x; t < SEGP * 4; t += 256) {
    const int seg = t >> 2, q4 = (t & 3) * 4;
    *(v4f*)(sA + seg * 16 + q4) = *(const v4f*)(gA + seg * 32 + q4);
    *(v4f*)(sB + seg * 16 + q4) = *(const v4f*)(gB + seg * 32 + q4);
  }
  __syncthreads();

  const int lane = threadIdx.x & 31;
  const int wave = threadIdx.x >> 5;
  const int half = lane >> 4;         // K-pair select for 16x4 f32 A/B fragments
  const int row = lane & 15;          // M for A-frag, N for B-frag

  float acc = 0.f;
  for (int t = wave; t < 36; t += 8) {          // 6x6 tiles of 16x16
    const int ti = (t * 171) >> 10;             // == t/6 for t in [0,36)
    const int tj = t - ti * 6;
    const int si = ti * 16 + row;
    const int sj = tj * 16 + row;
    const float* pa = sA + si * 16;
    const float* pb = sB + sj * 16;
    const int k0 = 2 * half;                    // 0 or 2
    v2f aN0 = {pa[0 + k0],  pa[1 + k0]};
    v2f aN1 = {pa[4 + k0],  pa[5 + k0]};
    v2f aD0 = {pa[8 + k0],  pa[9 + k0]};
    v2f aD1 = {pa[12 + k0], pa[13 + k0]};
    v2f bN0 = {pb[0 + k0],  pb[1 + k0]};
    v2f bN1 = {pb[4 + k0],  pb[5 + k0]};
    v2f bD0 = {pb[8 + k0],  pb[9 + k0]};
    v2f bD1 = {pb[12 + k0], pb[13 + k0]};

    v8f num = {};
    v8f dst = {};
    // D = A*B + C, 16x16 tile, K=4 each; two steps cover K=8 operand vectors
    num = __builtin_amdgcn_wmma_f32_16x16x4_f32(false, aN0, false, bN0, (short)0, num, false, false);
    num = __builtin_amdgcn_wmma_f32_16x16x4_f32(false, aN1, false, bN1, (short)0, num, false, false);
    dst = __builtin_amdgcn_wmma_f32_16x16x4_f32(false, aD0, false, bD0, (short)0, dst, false, false);
    dst = __builtin_amdgcn_wmma_f32_16x16x4_f32(false, aD1, false, bD1, (short)0, dst, false, false);

    #pragma unroll
    for (int r = 0; r < 8; ++r) {
      float d2 = fmaxf(dst[r], 0.f);                    // guard cancellation
      // den = dist^3 + eps ; raw v_sqrt_f32 / v_rcp_f32 (~1 ulp), well-
      // conditioned here: den >= 1e-8, num == 0 whenever d2 == 0 (pads).
      float den = d2 * __builtin_amdgcn_sqrtf(d2) + EPSV;
      acc = fmaf(num[r], __builtin_amdgcn_rcpf(den), acc);
    }
  }

  sRed[threadIdx.x] = acc;
  __syncthreads();
  for (int off = 128; off > 0; off >>= 1) {
    if (threadIdx.x < off) sRed[threadIdx.x] += sRed[threadIdx.x + off];
    __syncthreads();
  }
  if (threadIdx.x == 0) {
    const float v = sRed[0] * (1.0f / FOUR_PI);
    lk[((size_t)b * NCURVE + i) * NCURVE + j] = v;
    lk[((size_t)b * NCURVE + j) * NCURVE + i] = v;
  }
}

// ---------------------------------------------------------------------------
// Kernel 3: refined = lk*sigmoid(lk); mean/std(ddof=1); seed @ kernel; atan mean
// grid = BATCH blocks, 256 threads
// ---------------------------------------------------------------------------
__global__ void __launch_bounds__(256)
finalize_kernel(const float* __restrict__ lk, const float* __restrict__ ek,
                float* __restrict__ out) {
  const int b = blockIdx.x;
  const int tid = threadIdx.x;
  __shared__ float s1[256];
  __shared__ float s2[256];

  float sum = 0.f, sumsq = 0.f;
  for (int e = tid; e < NCURVE * NCURVE; e += 256) {
    float x = lk[(size_t)b * NCURVE * NCURVE + e];
    float sg = 1.f / (1.f + expf(-x));
    float r = x * sg;
    sum += r;
    sumsq += r * r;
  }
  s1[tid] = sum;
  s2[tid] = sumsq;
  __syncthreads();
  for (int off = 128; off > 0; off >>= 1) {
    if (tid < off) { s1[tid] += s1[tid + off]; s2[tid] += s2[tid + off]; }
    __syncthreads();
  }
  const float n = (float)(NCURVE * NCURVE);
  const float mu = s1[0] / n;
  const float var = fmaxf((s2[0] - n * mu * mu) / (n - 1.f), 0.f);
  const float sd = sqrtf(var);
  __syncthreads();   // done reading s1/s2 before reuse

  // l2_knot[b][tid] = mu*ek[0][tid] + sd*ek[1][tid]
  const float kk = mu * ek[tid] + sd * ek[LATENT + tid];
  out[(size_t)b * LATENT + tid] = kk;

  s1[tid] = atanf(kk);
  __syncthreads();
  for (int off = 128; off > 0; off >>= 1) {
    if (tid < off) s1[tid] += s1[tid + off];
    __syncthreads();
  }
  if (tid == 0)
    out[BATCH * LATENT + b] = s1[0] / ((float)LATENT * 3.14159265358979323846f);
}

// ---------------------------------------------------------------------------
extern "C" void kernel_launch(void* const* d_in, const int* in_sizes, int n_in,
                              void* d_out, int out_size, void* d_ws, size_t ws_size,
                              hipStream_t stream) {
  const float* q = (const float*)d_in[0];        // [2,32,96,4]
  const float* ek = (const float*)d_in[1];       // [2,256]
  float* out = (float*)d_out;                    // 514 floats
  float* ops = (float*)d_ws;                     // operand tensors
  float* lk = ops + OPS_TOTAL;                   // [2][32][32]

  prep_kernel<<<BATCH * NCURVE, SEGP, 0, stream>>>(q, ops);
  pair_kernel<<<BATCH * 528, 256, 0, stream>>>(ops, lk);
  finalize_kernel<<<BATCH, 256, 0, stream>>>(lk, ek, out);
}